// Glove2kSparse_20212116095162
// MI455X (gfx1250) — compile-verified
//
#include <hip/hip_runtime.h>
#include <float.h>

typedef float v2f __attribute__((ext_vector_type(2)));
typedef float v8f __attribute__((ext_vector_type(8)));

#define B_ROWS 131072
#define D_IN   100
#define D_CODE 1000
#define KSEL   25

#define BM   32      // batch rows per workgroup
#define NW   8       // waves per workgroup (wave32)
#define NTHR 256
#define CPAD 1024    // padded code dimension for selection
#define HP   1028    // h row pitch in dwords (conflict-free C-fragment stores)
#define XP   106     // x row pitch in dwords (conflict-free, 8B-aligned A reads)

// ---------------------------------------------------------------------------
// Tiny one-shot transpose: WdecT[c*100+o] = Wdec[o*1000+c]  (coalesced decoder)
// ---------------------------------------------------------------------------
__global__ void sae_transpose_wdec(const float* __restrict__ Wdec,
                                   float* __restrict__ WdecT) {
  int i = blockIdx.x * blockDim.x + threadIdx.x;
  if (i < D_IN * D_CODE) {
    int c = i / D_IN;
    int o = i - c * D_IN;
    WdecT[i] = Wdec[(size_t)o * D_CODE + c];
  }
}

// ---------------------------------------------------------------------------
// Fused: encoder (WMMA f32, dual accumulator chains) -> exact top-25
//        threshold (byte radix select) -> sparse decoder.
// ---------------------------------------------------------------------------
__global__ __launch_bounds__(NTHR) void sae_fused(
    const float* __restrict__ x,    const float* __restrict__ Wenc,
    const float* __restrict__ benc, const float* __restrict__ Wdec,
    const float* __restrict__ bdec, const float* __restrict__ WdecT,
    float* __restrict__ out, int useT)
{
  __shared__ float          xs[BM * XP];
  __shared__ float          hb[BM * HP];
  __shared__ unsigned       hist[NW][256];
  __shared__ unsigned       cnt[BM];
  __shared__ unsigned short lst[BM][32];
  __shared__ unsigned       selw[NW][2];
  __shared__ float          thr[BM];

  const int tid   = threadIdx.x;
  const int lane  = tid & 31;
  const int wave  = tid >> 5;
  const int rbase = blockIdx.x * BM;

  // ---- phase 1: stage x tile (non-temporal: read-once stream) --------------
  for (int i = tid; i < BM * D_IN; i += NTHR) {
    int r = i / D_IN, k = i - r * D_IN;
    xs[r * XP + k] =
        __builtin_nontemporal_load(&x[(size_t)(rbase + r) * D_IN + k]);
  }
  __syncthreads();

  // ---- phase 2: h = x @ Wenc^T + benc via v_wmma_f32_16x16x4_f32 -----------
  // Each wave computes two 16x16 N-tiles sharing one A fragment: two
  // independent WMMA accumulation chains -> hides XDL accumulate latency.
  // Pairs cover nt 0..63 (cols 0..1023): every padded h element is written.
  {
    const int m  = lane & 15;          // A: row / B,C: column-in-tile
    const int kb = (lane >> 4) << 1;   // K sub-offset per lane half
    for (int p = wave; p < 64; p += NW) {        // exactly 8 pairs per wave
      const int mt  = p & 1;
      const int nt0 = (p >> 1) * 2;
      const int nt1 = nt0 + 1;
      const int c0  = nt0 * 16 + m;
      const int c1  = nt1 * 16 + m;
      const int cs0 = (c0 < D_CODE) ? c0 : (D_CODE - 1);  // clamp OOB codes
      const int cs1 = (c1 < D_CODE) ? c1 : (D_CODE - 1);
      const float* wb0 = Wenc + (size_t)cs0 * D_IN + kb;  // B: Wenc[c][k]
      const float* wb1 = Wenc + (size_t)cs1 * D_IN + kb;
      const float* xa  = xs + m * XP + kb;                // A: x[m][k] (LDS)
      v8f acc0 = {0.f, 0.f, 0.f, 0.f, 0.f, 0.f, 0.f, 0.f};
      v8f acc1 = {0.f, 0.f, 0.f, 0.f, 0.f, 0.f, 0.f, 0.f};
#pragma unroll
      for (int ks = 0; ks < 25; ++ks) {           // K = 100 = 25 * 4
        v2f a  = *(const v2f*)(xa  + ks * 4);
        v2f b0 = *(const v2f*)(wb0 + ks * 4);
        v2f b1 = *(const v2f*)(wb1 + ks * 4);
        acc0 = __builtin_amdgcn_wmma_f32_16x16x4_f32(
            false, a, false, b0, (short)0, acc0, false, false);
        acc1 = __builtin_amdgcn_wmma_f32_16x16x4_f32(
            false, a, false, b1, (short)0, acc1, false, false);
      }
      const float bv0   = (c0 < D_CODE) ? benc[c0] : 0.0f;
      const float bv1   = (c1 < D_CODE) ? benc[c1] : 0.0f;
      const int   rhalf = (lane >> 4) << 3;       // C: lanes 16-31 hold M+8
#pragma unroll
      for (int r = 0; r < 8; ++r) {
        int row = mt * 16 + rhalf + r;
        hb[row * HP + nt0 * 16 + m] =
            (c0 < D_CODE) ? (acc0[r] + bv0) : -FLT_MAX;
        hb[row * HP + nt1 * 16 + m] =
            (c1 < D_CODE) ? (acc1[r] + bv1) : -FLT_MAX;
      }
    }
  }
  __syncthreads();

  // ---- phase 3: exact 25th-largest per row (4-pass byte radix select) ------
  for (int rr = 0; rr < BM / NW; ++rr) {          // 4 rows per wave, lockstep
    const int    r    = wave * (BM / NW) + rr;
    const float* hrow = hb + r * HP;
    unsigned prefix = 0, need = KSEL;
    for (int p = 3; p >= 0; --p) {
      const int shift = p * 8;
#pragma unroll
      for (int j = 0; j < 8; ++j) hist[wave][lane * 8 + j] = 0u;
      __syncthreads();
      for (int j = 0; j < 32; ++j) {              // lane scans 32 values
        float    f = hrow[lane + 32 * j];
        unsigned s = __float_as_uint(f);
        unsigned u = s ^ ((s & 0x80000000u) ? 0xFFFFFFFFu : 0x80000000u);
        bool match =
            (((unsigned long long)(u ^ prefix)) >> (shift + 8)) == 0ULL;
        if (match) atomicAdd(&hist[wave][(u >> shift) & 255u], 1u);
      }
      __syncthreads();
      unsigned v[8], s = 0;                       // descending suffix scan
#pragma unroll
      for (int j = 0; j < 8; ++j) { v[j] = hist[wave][255 - lane * 8 - j]; s += v[j]; }
      unsigned incl = s;
      for (int d = 1; d < 32; d <<= 1) {
        unsigned t = __shfl_up(incl, d);
        if (lane >= d) incl += t;
      }
      unsigned excl = incl - s;
      if (excl < need && need <= incl) {          // exactly one lane
        unsigned cum = excl;
        int j = 0;
        while (cum + v[j] < need) { cum += v[j]; ++j; }
        selw[wave][0] = (unsigned)(255 - lane * 8 - j);
        selw[wave][1] = need - cum;
      }
      __syncthreads();
      prefix |= selw[wave][0] << shift;
      need    = selw[wave][1];
      __syncthreads();
    }
    if (lane == 0) {                              // invert sortable mapping
      unsigned s = (prefix & 0x80000000u) ? (prefix ^ 0x80000000u) : ~prefix;
      thr[r] = __uint_as_float(s);
    }
  }
  __syncthreads();

  // ---- phase 4: build per-row nonzero index lists --------------------------
  if (tid < BM) cnt[tid] = 0u;
  __syncthreads();
  for (int i = tid; i < BM * CPAD; i += NTHR) {
    int   r = i >> 10, c = i & (CPAD - 1);
    float f = hb[r * HP + c];
    if (f >= thr[r]) {                            // pads are -FLT_MAX: excluded
      unsigned pos = atomicAdd(&cnt[r], 1u);
      if (pos < 32u) lst[r][pos] = (unsigned short)c;
    }
  }
  __syncthreads();

  // ---- phase 5: sparse decoder  out = h_sparse @ Wdec^T + bdec -------------
  for (int rr = 0; rr < BM / NW; ++rr) {
    const int r    = wave * (BM / NW) + rr;
    const int grow = rbase + r;
    unsigned  n    = cnt[r]; if (n > 32u) n = 32u;
    float acc[4];
#pragma unroll
    for (int i = 0; i < 4; ++i) {
      int o = lane + 32 * i;
      acc[i] = (o < D_IN) ? bdec[o] : 0.0f;
    }
    for (unsigned j = 0; j < n; ++j) {
      const int   c  = lst[r][j];
      const float hv = hb[r * HP + c];            // LDS broadcast read
#pragma unroll
      for (int i = 0; i < 4; ++i) {
        int o = lane + 32 * i;
        if (o < D_IN) {
          float w = useT ? WdecT[c * D_IN + o]
                         : Wdec[(size_t)o * D_CODE + c];
          acc[i] += hv * w;
        }
      }
    }
#pragma unroll
    for (int i = 0; i < 4; ++i) {
      int o = lane + 32 * i;
      if (o < D_IN)
        __builtin_nontemporal_store(acc[i], &out[(size_t)grow * D_IN + o]);
    }
  }
}

extern "C" void kernel_launch(void* const* d_in, const int* in_sizes, int n_in,
                              void* d_out, int out_size, void* d_ws, size_t ws_size,
                              hipStream_t stream) {
  const float* x    = (const float*)d_in[0];
  const float* Wenc = (const float*)d_in[1];
  const float* benc = (const float*)d_in[2];
  const float* Wdec = (const float*)d_in[3];
  const float* bdec = (const float*)d_in[4];
  float*       out  = (float*)d_out;
  float*       WdT  = (float*)d_ws;

  const size_t tbytes = (size_t)D_IN * D_CODE * sizeof(float);
  const int useT = (ws_size >= tbytes) ? 1 : 0;

  if (useT) {
    sae_transpose_wdec<<<(D_IN * D_CODE + 255) / 256, 256, 0, stream>>>(Wdec, WdT);
  }
  sae_fused<<<B_ROWS / BM, NTHR, 0, stream>>>(x, Wenc, benc, Wdec, bdec, WdT,
                                              out, useT);
}